// GNN_37014028156991
// MI455X (gfx1250) — compile-verified
//
#include <hip/hip_runtime.h>

typedef __attribute__((ext_vector_type(2))) float v2f;
typedef __attribute__((ext_vector_type(8))) float v8f;

#define NN   100000
#define NE   1000000
#define IND  128
#define HID  128
#define OUTD 64
#define NL   3
#define LN_EPS 1e-5f

// ---------------------------------------------------------------------------
// Core f32 WMMA tile accumulator: one wave computes a 16x16 f32 tile.
// A is row-major [M,K], pre-offset to this lane's row (row0 + (lane&15)).
// W is row-major [N,K] (PyTorch-style weight), pre-offset to this lane's
// output column row (col0 + (lane&15)); B = W^T so b[k][n] = W[n][k].
// ISA layout (32-bit A 16x4): lanes 0-15 hold K={0,1}, lanes 16-31 K={2,3}.
// ascale lets us fold the 1/deg row scaling into the A operand for free.
// ---------------------------------------------------------------------------
__device__ __forceinline__ v8f wmma_accum_f32(const float* __restrict__ Arow,
                                              const float* __restrict__ Wrow,
                                              float ascale, int K, v8f c) {
  const int lane = threadIdx.x & 31;
  const int kh = (lane >> 4) << 1;  // 0 for lanes 0-15, 2 for lanes 16-31
  for (int k0 = 0; k0 < K; k0 += 4) {
    v2f a, b;
    a.x = Arow[k0 + kh] * ascale;
    a.y = Arow[k0 + kh + 1] * ascale;
    b.x = Wrow[k0 + kh];
    b.y = Wrow[k0 + kh + 1];
    // (neg_a, A, neg_b, B, c_mod, C, reuse_a, reuse_b)
    c = __builtin_amdgcn_wmma_f32_16x16x4_f32(false, a, false, b, (short)0, c,
                                              false, false);
  }
  return c;
}

// C = A @ W^T + bias   (A:[M,K], W:[N,K], C:[M,N]); one wave per 16x16 tile.
__global__ void gemm_bias_kernel(const float* __restrict__ A,
                                 const float* __restrict__ W,
                                 const float* __restrict__ bias,
                                 float* __restrict__ C, int M, int N, int K) {
  const int lane = threadIdx.x & 31;
  const int m = lane & 15;
  const int row0 = blockIdx.x << 4;
  const int col0 = (threadIdx.x >> 5) << 4;
  if (row0 >= M || col0 >= N) return;  // wave-uniform: EXEC stays all-ones
  v8f c = {};
  c = wmma_accum_f32(A + (size_t)(row0 + m) * K, W + (size_t)(col0 + m) * K,
                     1.0f, K, c);
  const int rb = (lane >> 4) << 3;    // 0 or 8
  const float bn = bias[col0 + m];    // n == lane&15 == m
  for (int r = 0; r < 8; ++r)
    C[(size_t)(row0 + rb + r) * N + col0 + m] = c[r] + bn;
}

// h += (agg/deg) @ Wl^T + bl + u @ Wr^T   (fused dual-GEMM SAGE layer, N=K=HID)
__global__ void sage_layer_kernel(const float* __restrict__ agg,
                                  const float* __restrict__ deg,
                                  const float* __restrict__ u,
                                  const float* __restrict__ Wl,
                                  const float* __restrict__ bl,
                                  const float* __restrict__ Wr,
                                  float* __restrict__ h) {
  const int lane = threadIdx.x & 31;
  const int m = lane & 15;
  const int row0 = blockIdx.x << 4;
  const int col0 = (threadIdx.x >> 5) << 4;
  const float invd = 1.0f / fmaxf(deg[row0 + m], 1.0f);
  v8f c = {};
  c = wmma_accum_f32(agg + (size_t)(row0 + m) * HID,
                     Wl + (size_t)(col0 + m) * HID, invd, HID, c);
  c = wmma_accum_f32(u + (size_t)(row0 + m) * HID,
                     Wr + (size_t)(col0 + m) * HID, 1.0f, HID, c);
  const int rb = (lane >> 4) << 3;
  const float bn = bl[col0 + m];
  for (int r = 0; r < 8; ++r) {
    const size_t idx = (size_t)(row0 + rb + r) * HID + col0 + m;
    h[idx] = h[idx] + c[r] + bn;  // residual; each element owned by one thread
  }
}

// u = relu(layer_norm(h)); one wave32 per 128-wide row, float4 per lane.
__global__ void ln_relu_kernel(const float* __restrict__ h,
                               const float* __restrict__ g,
                               const float* __restrict__ b,
                               float* __restrict__ u, int M) {
  const int row = blockIdx.x * (blockDim.x >> 5) + (threadIdx.x >> 5);
  if (row >= M) return;
  const int lane = threadIdx.x & 31;
  const float4 v = reinterpret_cast<const float4*>(h + (size_t)row * HID)[lane];
  float s = v.x + v.y + v.z + v.w;
  for (int off = 16; off > 0; off >>= 1) s += __shfl_xor(s, off, 32);
  const float mu = s * (1.0f / HID);
  const float4 d = make_float4(v.x - mu, v.y - mu, v.z - mu, v.w - mu);
  float vs = d.x * d.x + d.y * d.y + d.z * d.z + d.w * d.w;
  for (int off = 16; off > 0; off >>= 1) vs += __shfl_xor(vs, off, 32);
  const float inv = rsqrtf(vs * (1.0f / HID) + LN_EPS);
  const float4 gg = reinterpret_cast<const float4*>(g)[lane];
  const float4 bb = reinterpret_cast<const float4*>(b)[lane];
  float4 o;
  o.x = fmaxf(d.x * inv * gg.x + bb.x, 0.0f);
  o.y = fmaxf(d.y * inv * gg.y + bb.y, 0.0f);
  o.z = fmaxf(d.z * inv * gg.z + bb.z, 0.0f);
  o.w = fmaxf(d.w * inv * gg.w + bb.w, 0.0f);
  reinterpret_cast<float4*>(u + (size_t)row * HID)[lane] = o;
}

__global__ void zero_kernel(float* __restrict__ p, int n) {
  const int i = blockIdx.x * blockDim.x + threadIdx.x;
  if (i < n) p[i] = 0.0f;
}

// One wave per edge: gather u[src] (L2-resident, 51MB table), atomic-add into
// agg[dst], and count degree. float4 per lane -> 128 floats per edge.
__global__ void scatter_kernel(const float* __restrict__ u,
                               const long long* __restrict__ src,
                               const long long* __restrict__ dst,
                               float* __restrict__ agg,
                               float* __restrict__ deg) {
  const int e = blockIdx.x * (blockDim.x >> 5) + (threadIdx.x >> 5);
  if (e >= NE) return;
  const int lane = threadIdx.x & 31;
  const int s = (int)src[e];
  const int d = (int)dst[e];
  const float4 mv = reinterpret_cast<const float4*>(u + (size_t)s * HID)[lane];
  float* ap = agg + (size_t)d * HID + lane * 4;
  atomicAdd(ap + 0, mv.x);
  atomicAdd(ap + 1, mv.y);
  atomicAdd(ap + 2, mv.z);
  atomicAdd(ap + 3, mv.w);
  if (lane == 0) atomicAdd(deg + d, 1.0f);
}

extern "C" void kernel_launch(void* const* d_in, const int* in_sizes, int n_in,
                              void* d_out, int out_size, void* d_ws,
                              size_t ws_size, hipStream_t stream) {
  const float* x          = (const float*)d_in[0];
  const long long* ei     = (const long long*)d_in[1];  // int64 in reference
  const float* lin1_w     = (const float*)d_in[2];
  const float* lin1_b     = (const float*)d_in[3];
  const float* ln_g       = (const float*)d_in[4];
  const float* ln_b       = (const float*)d_in[5];
  const float* Wl         = (const float*)d_in[6];
  const float* bl         = (const float*)d_in[7];
  const float* Wr         = (const float*)d_in[8];
  const float* nf_g       = (const float*)d_in[9];
  const float* nf_b       = (const float*)d_in[10];
  const float* lin2_w     = (const float*)d_in[11];
  const float* lin2_b     = (const float*)d_in[12];
  float* out = (float*)d_out;

  // workspace: h | u | agg | deg  (3*12.8M + 0.1M floats ~ 154 MB)
  float* h   = (float*)d_ws;
  float* u   = h + (size_t)NN * HID;
  float* agg = u + (size_t)NN * HID;
  float* deg = agg + (size_t)NN * HID;

  const long long* src = ei;        // edge_index[0]
  const long long* dst = ei + NE;   // edge_index[1]

  const int mtiles = NN / 16;       // 6250, exact

  // h = x @ lin1_w^T + lin1_b
  gemm_bias_kernel<<<mtiles, (HID / 16) * 32, 0, stream>>>(x, lin1_w, lin1_b,
                                                           h, NN, HID, IND);
  for (int l = 0; l < NL; ++l) {
    ln_relu_kernel<<<(NN + 7) / 8, 256, 0, stream>>>(
        h, ln_g + (size_t)l * HID, ln_b + (size_t)l * HID, u, NN);
    const int zn = NN * HID + NN;   // agg and deg are contiguous
    zero_kernel<<<(zn + 255) / 256, 256, 0, stream>>>(agg, zn);
    scatter_kernel<<<(NE + 7) / 8, 256, 0, stream>>>(u, src, dst, agg, deg);
    sage_layer_kernel<<<mtiles, (HID / 16) * 32, 0, stream>>>(
        agg, deg, u, Wl + (size_t)l * HID * HID, bl + (size_t)l * HID,
        Wr + (size_t)l * HID * HID, h);
  }
  ln_relu_kernel<<<(NN + 7) / 8, 256, 0, stream>>>(h, nf_g, nf_b, u, NN);
  // out = u @ lin2_w^T + lin2_b
  gemm_bias_kernel<<<mtiles, (OUTD / 16) * 32, 0, stream>>>(
      u, lin2_w, lin2_b, out, NN, OUTD, HID);
}